// SoftAUCLoss_6768868459297
// MI455X (gfx1250) — compile-verified
//
#include <hip/hip_runtime.h>
#include <hip/hip_bf16.h>

typedef __attribute__((ext_vector_type(2))) float v2f;
typedef __attribute__((ext_vector_type(8))) float v8f;

#define GAMMA_F 0.2f
#define NBIG    1.0e30f
#define TILE    16
#define BLOCK   256
#define MAXN    16384

// ---------------------------------------------------------------------------
// Kernel 1: per-element prep.
//   s_i = softmax(y_pred_i)[1] = sigmoid(p1 - p0)
//   a[j] = 0.5 * (neg_j ?  s_j        : -BIG)   (row operand, negatives)
//   b[i] = 0.5 * (pos_i ? (gamma-s_i) : -BIG)   (col operand, positives)
// Masks folded into values: masked entries drive the hinge to 0. Pre-halved
// because the palindromic WMMA formulation doubles the sum.
// Also zeroes the output accumulator (harness poisons d_out with 0xAA).
// ---------------------------------------------------------------------------
__global__ void soft_auc_prep(const float* __restrict__ y_pred,
                              const int*   __restrict__ y_true,
                              float* __restrict__ a_half,
                              float* __restrict__ b_half,
                              float* __restrict__ out,
                              int n) {
    int i = blockIdx.x * blockDim.x + threadIdx.x;
    if (i == 0) out[0] = 0.0f;
    if (i >= n) return;
    float p0 = y_pred[2 * i + 0];
    float p1 = y_pred[2 * i + 1];
    float s  = 1.0f / (1.0f + __expf(p0 - p1));   // prob of class 1
    bool  pos = (y_true[i] == 1);
    a_half[i] = 0.5f * (pos ? -NBIG : s);
    b_half[i] = 0.5f * (pos ? (GAMMA_F - s) : -NBIG);
}

// ---------------------------------------------------------------------------
// Kernel 2: tiled pairwise hinge^3 via V_WMMA_F32_16X16X4_F32.
//
// Rank-2 outer product per 16x16 tile (palindromic in K, so the packing is
// invariant to the half-wave K ordering of the 4x16 B layout):
//   A row m = (a[m], 1, 1, a[m]),  B col n = (1, b[n], b[n], 1)
//   => D[m][n] = 2*(a_half[m] + b_half[n]) = s_j + (gamma - s_i)
// C/D permutation is irrelevant: the whole tile is reduced.
// Epilogue per element: t = med3(d, 0, +MAX)  [1-op relu, no canonicalize],
// acc += t^3 via mul+fmac, 4-way split accumulators for FMA ILP.
// ---------------------------------------------------------------------------
__global__ void __launch_bounds__(BLOCK)
soft_auc_pairs(const float* __restrict__ a_half,
               const float* __restrict__ b_half,
               float* __restrict__ out,
               int n) {
    __shared__ float bsh[MAXN];              // 64 KB of the 320 KB/WGP LDS

    const int tid = threadIdx.x;
    for (int idx = tid; idx < n; idx += BLOCK)   // cooperative stage to LDS
        bsh[idx] = b_half[idx];
    __syncthreads();

    const int  lane = tid & 31;
    const int  wave = tid >> 5;
    const int  m0   = (blockIdx.x * (BLOCK / 32) + wave) * TILE;
    const bool lo   = (lane < 16);
    const int  r    = lane & 15;

    float acc0 = 0.0f, acc1 = 0.0f, acc2 = 0.0f, acc3 = 0.0f;
    if (m0 < n) {
        // A operand: 16x4 f32. lanes 0-15 hold (K0,K1)=(a,1) in (v0,v1);
        // lanes 16-31 hold (K2,K3)=(1,a). Palindromic rows (a,1,1,a).
        const float am = a_half[m0 + r];
        v2f A;
        A.x = lo ? am   : 1.0f;
        A.y = lo ? 1.0f : am;

        const float big = 3.0e38f;           // finite upper clamp for med3-relu
        for (int n0 = 0; n0 < n; n0 += TILE) {
            const float bv = bsh[n0 + r];
            // B operand: 4x16 f32, palindromic rows (1,b,b,1).
            v2f B;
            B.x = lo ? 1.0f : bv;
            B.y = lo ? bv   : 1.0f;
            v8f C = {};
            // D[m][n] = a[m] + b[n]  (pre-halved operands cancel the 2x)
            v8f D = __builtin_amdgcn_wmma_f32_16x16x4_f32(
                /*neg_a=*/false, A, /*neg_b=*/false, B,
                /*c_mod=*/(short)0, C, /*reuse_a=*/false, /*reuse_b=*/false);

            // relu(d)^3 with 3 VALU/element: v_med3_f32 + v_mul + v_fmac.
            // All D entries are finite (masks are -1e30), so med3 == relu.
            float t0 = __builtin_amdgcn_fmed3f(D[0], 0.0f, big);
            float t1 = __builtin_amdgcn_fmed3f(D[1], 0.0f, big);
            float t2 = __builtin_amdgcn_fmed3f(D[2], 0.0f, big);
            float t3 = __builtin_amdgcn_fmed3f(D[3], 0.0f, big);
            float t4 = __builtin_amdgcn_fmed3f(D[4], 0.0f, big);
            float t5 = __builtin_amdgcn_fmed3f(D[5], 0.0f, big);
            float t6 = __builtin_amdgcn_fmed3f(D[6], 0.0f, big);
            float t7 = __builtin_amdgcn_fmed3f(D[7], 0.0f, big);
            acc0 = __builtin_fmaf(t0 * t0, t0, acc0);
            acc1 = __builtin_fmaf(t1 * t1, t1, acc1);
            acc2 = __builtin_fmaf(t2 * t2, t2, acc2);
            acc3 = __builtin_fmaf(t3 * t3, t3, acc3);
            acc0 = __builtin_fmaf(t4 * t4, t4, acc0);
            acc1 = __builtin_fmaf(t5 * t5, t5, acc1);
            acc2 = __builtin_fmaf(t6 * t6, t6, acc2);
            acc3 = __builtin_fmaf(t7 * t7, t7, acc3);
        }
    }

    float acc = (acc0 + acc1) + (acc2 + acc3);
    // wave32 tree reduction, then one f32 atomic per wave
#pragma unroll
    for (int off = 16; off > 0; off >>= 1)
        acc += __shfl_down(acc, off, 32);
    if (lane == 0)
        atomicAdd(out, acc);
}

// ---------------------------------------------------------------------------
extern "C" void kernel_launch(void* const* d_in, const int* in_sizes, int n_in,
                              void* d_out, int out_size, void* d_ws, size_t ws_size,
                              hipStream_t stream) {
    (void)n_in; (void)out_size; (void)ws_size;

    const float* y_pred = (const float*)d_in[0];   // [N,2] f32
    const int*   y_true = (const int*)d_in[1];     // [N]   i32
    float*       out    = (float*)d_out;           // scalar f32

    const int n = in_sizes[1];                     // N = 16384

    float* a_half = (float*)d_ws;                  // [N]
    float* b_half = a_half + n;                    // [N]  (128 KB total)

    // Pass 1: sigmoid + mask-fold + output zeroing
    soft_auc_prep<<<(n + BLOCK - 1) / BLOCK, BLOCK, 0, stream>>>(
        y_pred, y_true, a_half, b_half, out, n);

    // Pass 2: one wave per 16-row strip, 8 waves per block
    const int strips = (n + TILE - 1) / TILE;          // 1024
    const int wavesPerBlock = BLOCK / 32;              // 8
    const int blocks = (strips + wavesPerBlock - 1) / wavesPerBlock;  // 128
    soft_auc_pairs<<<blocks, BLOCK, 0, stream>>>(a_half, b_half, out, n);
}